// MTAD_GAT_37941741093077
// MI455X (gfx1250) — compile-verified
//
#include <hip/hip_runtime.h>
#include <hip/hip_bf16.h>

// ---------------------------------------------------------------------------
// MTAD-GAT forward for MI455X (gfx1250), fp32 via V_WMMA_F32_16X16X4_F32.
// ---------------------------------------------------------------------------

typedef __attribute__((ext_vector_type(2))) float v2f;
typedef __attribute__((ext_vector_type(8))) float v8f;

#define NBATCH 16
#define WDIM   128
#define KDIM   128
#define KSZ    7
#define HGD    256
#define EDM    256
#define G3     768
#define HCATD  384   // xconv | h_feat | h_temp fused along channels
#define LRELU_ALPHA 0.2f

static __device__ __forceinline__ v8f wmma_f32_16x16x4(v2f a, v2f b, v8f c) {
  // 8 args: (neg_a, A, neg_b, B, c_mod, C, reuse_a, reuse_b)
  return __builtin_amdgcn_wmma_f32_16x16x4_f32(false, a, false, b, (short)0, c,
                                               false, false);
}

static __device__ __forceinline__ float sigmoid_f(float x) {
  return 1.0f / (1.0f + __expf(-x));
}

// ---------------------------------------------------------------------------
// Generic strided GEMM: C[r,e] (+)= sum_d A[r,d] * B[d,e] (+ bias[e]), act.
//   A element  : A[bb*a_bs + r*a_rs + d*a_cs]        (ACONT => a_cs==1)
//   B element  : Bm[bb*b_bs + b_off + e*b_es + d*b_ds] (BCONT => b_ds==1)
//   C element  : C[bb*c_bs + r*c_rs + e*c_cs]
// Block = 128 threads = 4 wave32s, each wave owns one 16x16 tile
// (4 consecutive e-tiles per block). act: 0=none, 1=relu, 2=sigmoid.
// ---------------------------------------------------------------------------
template <int ACONT, int BCONT>
__global__ void __launch_bounds__(128) gemm_wmma_kernel(
    const float* __restrict__ A, int a_rs, int a_cs, int a_bs,
    const float* __restrict__ Bm, int b_es, int b_ds, int b_bs, int b_off,
    int D,
    const float* __restrict__ bias,
    float* __restrict__ C, int c_rs, int c_cs, int c_bs,
    int accumulate, int act) {
  const int lane = threadIdx.x & 31;
  const int wave = threadIdx.x >> 5;  // 0..3
  const int hl = lane >> 4;           // half-wave select
  const int l = lane & 15;
  const int e0 = (blockIdx.x * 4 + wave) * 16;
  const int r0 = blockIdx.y * 16;
  const int bb = blockIdx.z;

  const float* Ab = A + (size_t)bb * a_bs;
  const float* Bb = Bm + (size_t)bb * b_bs + b_off;
  float* Cb = C + (size_t)bb * c_bs;

  const int row = r0 + l;
  const int col = e0 + l;
  const float* arow = Ab + (size_t)row * a_rs;
  const float* brow = Bb + (size_t)col * b_es;

  v8f acc = {};
  for (int d0 = 0; d0 < D; d0 += 4) {
    const int dk = d0 + 2 * hl;  // lanes 0-15 -> K={d0,d0+1}; 16-31 -> +2
    v2f av, bv;
    if (ACONT) {
      av = *(const v2f*)(arow + dk);  // 8B-aligned (dk even, strides even)
    } else {
      const float* ap = arow + (size_t)dk * a_cs;
      av[0] = ap[0];
      av[1] = ap[a_cs];
    }
    if (BCONT) {
      bv = *(const v2f*)(brow + dk);
    } else {
      const float* bp = brow + (size_t)dk * b_ds;
      bv[0] = bp[0];
      bv[1] = bp[b_ds];
    }
    acc = wmma_f32_16x16x4(av, bv, acc);
  }
  const float badd = bias ? bias[col] : 0.0f;
#pragma unroll
  for (int g = 0; g < 8; ++g) {
    const int m = g + 8 * hl;  // output row within tile
    const size_t idx = (size_t)(r0 + m) * c_rs + (size_t)col * c_cs;
    float v = acc[g] + badd;
    if (accumulate) v += Cb[idx];
    if (act == 1) v = fmaxf(v, 0.0f);
    else if (act == 2) v = sigmoid_f(v);
    Cb[idx] = v;
  }
}

static void launch_gemm(hipStream_t stream, int E, int Rtiles16, int NBz,
                        const float* A, int a_rs, int a_cs, int a_bs,
                        const float* Bm, int b_es, int b_ds, int b_bs,
                        int b_off, int D, const float* bias, float* C,
                        int c_rs, int c_cs, int c_bs, int acc, int act) {
  const dim3 grid(E / 64, Rtiles16, NBz);
  const dim3 blk(128);
  if (a_cs == 1 && b_ds == 1)
    gemm_wmma_kernel<1, 1><<<grid, blk, 0, stream>>>(
        A, a_rs, a_cs, a_bs, Bm, b_es, b_ds, b_bs, b_off, D, bias, C, c_rs,
        c_cs, c_bs, acc, act);
  else if (a_cs == 1)
    gemm_wmma_kernel<1, 0><<<grid, blk, 0, stream>>>(
        A, a_rs, a_cs, a_bs, Bm, b_es, b_ds, b_bs, b_off, D, bias, C, c_rs,
        c_cs, c_bs, acc, act);
  else if (b_ds == 1)
    gemm_wmma_kernel<0, 1><<<grid, blk, 0, stream>>>(
        A, a_rs, a_cs, a_bs, Bm, b_es, b_ds, b_bs, b_off, D, bias, C, c_rs,
        c_cs, c_bs, acc, act);
  else
    gemm_wmma_kernel<0, 0><<<grid, blk, 0, stream>>>(
        A, a_rs, a_cs, a_bs, Bm, b_es, b_ds, b_bs, b_off, D, bias, C, c_rs,
        c_cs, c_bs, acc, act);
}

// ---------------------------------------------------------------------------
// Conv1d (same padding, KS=7) + bias + ReLU, as 7 accumulated K-GEMMs.
// x: (B, W, K) -> writes channels [0,128) of HCAT (B, W, 384).
// Branchless padding: clamp row address, scale fragment by 0/1 mask
// (keeps EXEC uniform -> no saveexec/branch around loads, WMMA-friendly).
// ---------------------------------------------------------------------------
__global__ void __launch_bounds__(128)
conv_relu_wmma_kernel(const float* __restrict__ x, const float* __restrict__ w,
                      const float* __restrict__ bias,
                      float* __restrict__ hcat) {
  const int lane = threadIdx.x & 31;
  const int wave = threadIdx.x >> 5;
  const int hl = lane >> 4;
  const int l = lane & 15;
  const int k0 = (blockIdx.x * 4 + wave) * 16;
  const int w0 = blockIdx.y * 16;
  const int b = blockIdx.z;

  v8f acc = {};
  for (int t = 0; t < KSZ; ++t) {
    const int wrow = w0 + l + t - (KSZ - 1) / 2;
    const int wcl = min(max(wrow, 0), WDIM - 1);
    const float fm = (wrow == wcl) ? 1.0f : 0.0f;  // zero out-of-range taps
    const float* xp = x + ((size_t)b * WDIM + wcl) * KDIM;
    const float* wp = w + (size_t)(k0 + l) * (KDIM * KSZ) + t;
    for (int d0 = 0; d0 < KDIM; d0 += 4) {
      const int dk = d0 + 2 * hl;
      const v2f tv = *(const v2f*)(xp + dk);
      v2f av, bv;
      av[0] = tv[0] * fm;
      av[1] = tv[1] * fm;
      bv[0] = wp[(size_t)dk * KSZ];
      bv[1] = wp[(size_t)(dk + 1) * KSZ];
      acc = wmma_f32_16x16x4(av, bv, acc);
    }
  }
  const float badd = bias[k0 + l];
#pragma unroll
  for (int g = 0; g < 8; ++g) {
    const int m = g + 8 * hl;
    hcat[((size_t)b * WDIM + (w0 + m)) * HCATD + (k0 + l)] =
        fmaxf(acc[g] + badd, 0.0f);
  }
}

// ---------------------------------------------------------------------------
// GATv2 edge scores + softmax (factored form):
//   e[i,j] = sum_e a[e]*leaky(left[i,e]+rightT[e,j]) + bias[i,j]
//   attn[i,:] = softmax_j(e[i,:])
// rightT is stored (E, N) by the projection GEMM -> coalesced lane reads.
// Grid: (N=128, B), block 128 (one thread per j).
// ---------------------------------------------------------------------------
__global__ void __launch_bounds__(128)
gat_edge_softmax_kernel(const float* __restrict__ left,
                        const float* __restrict__ rightT,
                        const float* __restrict__ avec,
                        const float* __restrict__ ebias,
                        float* __restrict__ attn) {
  const int i = blockIdx.x;
  const int b = blockIdx.y;
  const int j = threadIdx.x;  // 0..127

  __shared__ float a_s[EDM];
  __shared__ float li_s[EDM];
  __shared__ float red[128];

  for (int e = j; e < EDM; e += 128) {
    a_s[e] = avec[e];
    li_s[e] = left[((size_t)b * 128 + i) * EDM + e];
  }
  __syncthreads();

  const float* rt = rightT + (size_t)b * 128 * EDM;  // [e*128 + j]
  float acc = 0.0f;
  for (int e = 0; e < EDM; ++e) {
    float h = li_s[e] + rt[e * 128 + j];
    h = (h > 0.0f) ? h : (LRELU_ALPHA * h);
    acc = fmaf(a_s[e], h, acc);
  }
  acc += ebias[i * 128 + j];

  red[j] = acc;
  __syncthreads();
  for (int s = 64; s > 0; s >>= 1) {
    if (j < s) red[j] = fmaxf(red[j], red[j + s]);
    __syncthreads();
  }
  const float mx = red[0];
  __syncthreads();
  const float ex = __expf(acc - mx);
  red[j] = ex;
  __syncthreads();
  for (int s = 64; s > 0; s >>= 1) {
    if (j < s) red[j] += red[j + s];
    __syncthreads();
  }
  attn[((size_t)b * 128 + i) * 128 + j] = ex / red[0];
}

// ---------------------------------------------------------------------------
// Sequential GRU, one persistent 512-thread workgroup (16 wave32s).
// Wave w owns gate columns c in [16w, 16w+16): it computes the r-block
// (rows c), z-block (rows 256+c) and n-block (rows 512+c) of gh = h@w_hh^T
// with THREE register-resident WMMA accumulators sharing one A fragment,
// then applies the gate nonlinearity locally -- no gh LDS round trip.
// Only h (16x256, 16 KB) lives in LDS. GI_CONST=1 (decoder): the repeated
// input's gi values are preloaded into registers once.
// ---------------------------------------------------------------------------
template <int GI_CONST>
__global__ void __launch_bounds__(512)
gru_seq_kernel(const float* __restrict__ gi, int gi_t_stride, int gi_m_stride,
               const float* __restrict__ w_hh, const float* __restrict__ b_hh,
               float* __restrict__ out_all, float* __restrict__ h_end, int T) {
  __shared__ float h_s[16 * HGD];  // 16 KB
  const int tid = threadIdx.x;
  const int lane = tid & 31, hl = lane >> 4, l = lane & 15;
  const int wave = tid >> 5;       // 0..15
  const int c = wave * 16 + l;     // this lane's gate column (0..255)

  const float* wr = w_hh + (size_t)c * HGD;              // r rows
  const float* wz = w_hh + (size_t)(c + HGD) * HGD;      // z rows
  const float* wn = w_hh + (size_t)(c + 2 * HGD) * HGD;  // n rows
  const float bhr = b_hh[c];
  const float bhz = b_hh[HGD + c];
  const float bhn = b_hh[2 * HGD + c];
  __builtin_prefetch(wr, 0, 0);  // warm L2/WGP$ (rows reused all T steps)
  __builtin_prefetch(wz, 0, 0);
  __builtin_prefetch(wn, 0, 0);

  float gir[8], giz[8], gin[8];
  if (GI_CONST) {
#pragma unroll
    for (int g = 0; g < 8; ++g) {
      const int m = g + 8 * hl;
      const float* gim = gi + (size_t)m * gi_m_stride;
      gir[g] = gim[c];
      giz[g] = gim[HGD + c];
      gin[g] = gim[2 * HGD + c];
    }
  }

  for (int idx = tid; idx < 16 * HGD; idx += 512) h_s[idx] = 0.0f;
  __syncthreads();

  for (int t = 0; t < T; ++t) {
    v8f ar = {}, az = {}, an = {};
    for (int d0 = 0; d0 < HGD; d0 += 4) {
      const int dk = d0 + 2 * hl;
      const v2f av = *(const v2f*)&h_s[l * HGD + dk];  // one ds_load_b64 ...
      const v2f br = *(const v2f*)(wr + dk);
      const v2f bz = *(const v2f*)(wz + dk);
      const v2f bn = *(const v2f*)(wn + dk);
      ar = wmma_f32_16x16x4(av, br, ar);  // ... feeds three WMMAs
      az = wmma_f32_16x16x4(av, bz, az);
      an = wmma_f32_16x16x4(av, bn, an);
    }
    __syncthreads();  // all waves done reading h_s before we overwrite it

#pragma unroll
    for (int g = 0; g < 8; ++g) {
      const int m = g + 8 * hl;  // batch row
      float gr_, gz_, gn_;
      if (GI_CONST) {
        gr_ = gir[g];
        gz_ = giz[g];
        gn_ = gin[g];
      } else {
        const float* gim =
            gi + (size_t)m * gi_m_stride + (size_t)t * gi_t_stride;
        gr_ = gim[c];
        gz_ = gim[HGD + c];
        gn_ = gim[2 * HGD + c];
      }
      const float r = sigmoid_f(gr_ + ar[g] + bhr);
      const float z = sigmoid_f(gz_ + az[g] + bhz);
      const float n = tanhf(gn_ + r * (an[g] + bhn));
      const float hprev = h_s[m * HGD + c];
      const float hnew = (1.0f - z) * n + z * hprev;
      h_s[m * HGD + c] = hnew;
      if (out_all) out_all[((size_t)m * T + t) * HGD + c] = hnew;
    }
    __syncthreads();
  }

  if (h_end)
    for (int idx = tid; idx < 16 * HGD; idx += 512) h_end[idx] = h_s[idx];
}

// ---------------------------------------------------------------------------
extern "C" void kernel_launch(void* const* d_in, const int* in_sizes, int n_in,
                              void* d_out, int out_size, void* d_ws,
                              size_t ws_size, hipStream_t stream) {
  const float* x        = (const float*)d_in[0];
  const float* conv_w   = (const float*)d_in[1];
  const float* conv_b   = (const float*)d_in[2];
  const float* fg_lin_w = (const float*)d_in[3];
  const float* fg_lin_b = (const float*)d_in[4];
  const float* fg_a     = (const float*)d_in[5];
  const float* fg_bias  = (const float*)d_in[6];
  const float* tg_lin_w = (const float*)d_in[7];
  const float* tg_lin_b = (const float*)d_in[8];
  const float* tg_a     = (const float*)d_in[9];
  const float* tg_bias  = (const float*)d_in[10];
  const float* gru_w_ih = (const float*)d_in[11];
  const float* gru_w_hh = (const float*)d_in[12];
  const float* gru_b_ih = (const float*)d_in[13];
  const float* gru_b_hh = (const float*)d_in[14];
  const float* fc1_w    = (const float*)d_in[15];
  const float* fc1_b    = (const float*)d_in[16];
  const float* fc2_w    = (const float*)d_in[17];
  const float* fc2_b    = (const float*)d_in[18];
  const float* dec_w_ih = (const float*)d_in[19];
  const float* dec_w_hh = (const float*)d_in[20];
  const float* dec_b_ih = (const float*)d_in[21];
  const float* dec_b_hh = (const float*)d_in[22];
  const float* rec_fc_w = (const float*)d_in[23];
  const float* rec_fc_b = (const float*)d_in[24];

  float* out = (float*)d_out;  // pred (16*128) ++ recons (16*128*128)

  // workspace layout (floats)
  float* ws = (float*)d_ws;
  float* HCAT  = ws;                  // (B,W,384) = 786432: xconv|hfeat|htemp
  float* LBUF  = HCAT + 786432;       // (B,128,256) = 524288
  float* RBUF  = LBUF + 524288;       // 524288 (stored transposed (E,N))
  float* ATTN  = RBUF + 524288;       // (B,128,128) = 262144
  float* GIALL = ATTN + 262144;       // (B,W,768) = 1572864
  float* HEND  = GIALL + 1572864;     // 16*256 = 4096
  float* DECGI = HEND + 4096;         // 16*768 = 12288
  float* DECOUT = DECGI + 12288;      // (B,W,256) = 524288
  float* FC1   = DECOUT + 524288;     // 16*256 = 4096

  const int HB = WDIM * HCATD;  // 49152, HCAT batch stride

  // 1) conv + relu -> HCAT[:, :, 0:128]
  conv_relu_wmma_kernel<<<dim3(2, 8, NBATCH), dim3(128), 0, stream>>>(
      x, conv_w, conv_b, HCAT);

  // 2) feature GAT (nodes = features, D = W): v[b,n,d] = HCAT[b,d,n]
  launch_gemm(stream, 256, 8, NBATCH, HCAT, 1, HCATD, HB,
              fg_lin_w, 256, 1, 0, 0, 128, fg_lin_b,
              LBUF, 256, 1, 32768, 0, 0);
  launch_gemm(stream, 256, 8, NBATCH, HCAT, 1, HCATD, HB,
              fg_lin_w, 256, 1, 0, 128, 128, nullptr,
              RBUF, 1, 128, 32768, 0, 0);  // transposed store (E,N)
  gat_edge_softmax_kernel<<<dim3(128, NBATCH), dim3(128), 0, stream>>>(
      LBUF, RBUF, fg_a, fg_bias, ATTN);
  // sigmoid(attn @ v), transposed store -> HCAT[:, :, 128:256]
  launch_gemm(stream, 128, 8, NBATCH, ATTN, 128, 1, 16384,
              HCAT, HCATD, 1, HB, 0, 128, nullptr,
              HCAT + 128, 1, HCATD, HB, 0, 2);

  // 3) temporal GAT (nodes = timesteps, D = K): v = HCAT[:, :, 0:128]
  launch_gemm(stream, 256, 8, NBATCH, HCAT, HCATD, 1, HB,
              tg_lin_w, 256, 1, 0, 0, 128, tg_lin_b,
              LBUF, 256, 1, 32768, 0, 0);
  launch_gemm(stream, 256, 8, NBATCH, HCAT, HCATD, 1, HB,
              tg_lin_w, 256, 1, 0, 128, 128, nullptr,
              RBUF, 1, 128, 32768, 0, 0);
  gat_edge_softmax_kernel<<<dim3(128, NBATCH), dim3(128), 0, stream>>>(
      LBUF, RBUF, tg_a, tg_bias, ATTN);
  launch_gemm(stream, 128, 8, NBATCH, ATTN, 128, 1, 16384,
              HCAT, 1, HCATD, HB, 0, 128, nullptr,
              HCAT + 256, HCATD, 1, HB, 0, 2);

  // 4) single batched input-side GRU GEMM: gi = HCAT @ w_ih.T + b_ih
  launch_gemm(stream, 768, 8, NBATCH, HCAT, HCATD, 1, HB,
              gru_w_ih, HCATD, 1, 0, 0, HCATD, gru_b_ih,
              GIALL, 768, 1, 98304, 0, 0);

  // 5) GRU encoder (sequential, persistent-LDS) -> h_end
  gru_seq_kernel<0><<<dim3(1), dim3(512), 0, stream>>>(
      GIALL, 768, 98304, gru_w_hh, gru_b_hh, nullptr, HEND, WDIM);

  // 6) decoder input GEMM once (input repeats every step)
  launch_gemm(stream, 768, 1, 1, HEND, 256, 1, 0,
              dec_w_ih, 256, 1, 0, 0, 256, dec_b_ih,
              DECGI, 768, 1, 0, 0, 0);

  // 7) GRU decoder (repeated input -> gi in registers) -> dec_out (b,t,HR)
  gru_seq_kernel<1><<<dim3(1), dim3(512), 0, stream>>>(
      DECGI, 0, 768, dec_w_hh, dec_b_hh, DECOUT, nullptr, WDIM);

  // 8) forecasting MLP -> pred = out[0 .. 2048)
  launch_gemm(stream, 256, 1, 1, HEND, 256, 1, 0,
              fc1_w, 256, 1, 0, 0, 256, fc1_b,
              FC1, 256, 1, 0, 0, 1);
  launch_gemm(stream, 128, 1, 1, FC1, 256, 1, 0,
              fc2_w, 256, 1, 0, 0, 256, fc2_b,
              out, 128, 1, 0, 0, 0);

  // 9) reconstruction head -> recons = out[2048 ..)
  launch_gemm(stream, 128, 8, NBATCH, DECOUT, 256, 1, 32768,
              rec_fc_w, 256, 1, 0, 0, 256, rec_fc_b,
              out + 2048, 128, 1, 16384, 0, 0);
}